// KohaNetwork_85907935854886
// MI455X (gfx1250) — compile-verified
//
#include <hip/hip_runtime.h>
#include <math.h>

typedef __attribute__((ext_vector_type(2))) float v2f;
typedef __attribute__((ext_vector_type(8))) float v8f;

#define BATCH   32
#define EMB     512
#define CONTEXT 64
#define RF      16
#define LCOLS   79              // CONTEXT + RF - 1
#define NSB     (EMB * LCOLS)  // 40448 floats per batch row of network_state
#define WBLK    (EMB * EMB)    // 262144 floats per per-block weight matrix
#define QBLK    (BATCH * EMB)  // 16384 floats per block for q/g/zbar

__device__ __forceinline__ v8f wmma4(v2f a, v2f b, v8f c) {
    // D = A(16x4,f32) * B(4x16,f32) + C(16x16,f32)
    return __builtin_amdgcn_wmma_f32_16x16x4_f32(
        false, a, false, b, (short)0, c, false, false);
}

// ---------------------------------------------------------------------------
// Kernel 0: out = network_state (cols >= CONTEXT survive; 0..63 overwritten
// later by k_out). Simple full copy keeps the tail handling trivial.
// ---------------------------------------------------------------------------
__global__ void k_copy(const float* __restrict__ ns, float* __restrict__ out, int n) {
    int i = blockIdx.x * blockDim.x + threadIdx.x;
    if (i < n) out[i] = ns[i];
}

// ---------------------------------------------------------------------------
// Kernel 1: q[c][b][e] = sum_d X[c][b][d] * Wq[c][d][e]
//   X[0][b][d]   = emb[idx[b]][d]
//   X[c>0][b][d] = ns[b][d][c-1]   (stride-79 gather over d)
// One wave -> 32x16 output slab. grid (64, 8), 128 thr (4 waves).
// ---------------------------------------------------------------------------
__global__ __launch_bounds__(128) void k_qproj(const int* __restrict__ idx,
                                               const float* __restrict__ ns,
                                               const float* __restrict__ emb,
                                               const float* __restrict__ Wq,
                                               float* __restrict__ qws) {
    const int c    = blockIdx.x;
    const int wave = threadIdx.x >> 5;
    const int lane = threadIdx.x & 31;
    const int n0   = (blockIdx.y * 4 + wave) * 16;
    const int klo  = (lane >> 4) << 1;   // 0 (lanes 0-15) or 2 (lanes 16-31)
    const int mlo  = lane & 15;

    const float* a0p;
    const float* a1p;
    int astep;
    if (c == 0) {
        a0p = emb + (size_t)idx[mlo] * EMB;
        a1p = emb + (size_t)idx[mlo + 16] * EMB;
        astep = 1;
    } else {
        a0p = ns + (size_t)mlo * NSB + (c - 1);
        a1p = ns + (size_t)(mlo + 16) * NSB + (c - 1);
        astep = LCOLS;
    }
    a0p += (size_t)klo * astep;
    a1p += (size_t)klo * astep;
    const float* bp = Wq + (size_t)c * WBLK + (size_t)klo * EMB + n0 + mlo;

    v8f acc0 = {}; v8f acc1 = {};
#pragma unroll 4
    for (int kk = 0; kk < EMB; kk += 4) {
        v2f a0; a0.x = a0p[0]; a0.y = a0p[astep];
        v2f a1; a1.x = a1p[0]; a1.y = a1p[astep];
        v2f b;  b.x  = bp[0];  b.y  = bp[EMB];
        acc0 = wmma4(a0, b, acc0);
        acc1 = wmma4(a1, b, acc1);
        a0p += 4 * astep; a1p += 4 * astep; bp += 4 * EMB;
    }

    float* q0 = qws + (size_t)c * QBLK;
    const int e   = n0 + mlo;
    const int bhi = (lane >> 4) * 8;
#pragma unroll
    for (int i = 0; i < 8; ++i) {
        q0[(size_t)(i + bhi) * EMB + e]      = acc0[i];
        q0[(size_t)(i + bhi + 16) * EMB + e] = acc1[i];
    }
}

// ---------------------------------------------------------------------------
// Kernel 2: g[c][b][d] = sum_e q[c][b][e] * Wk[c][d][e]   (q @ Wk^T)
// B[k=e][n=d] = Wk[d*512+e] -> per-lane rows of Wk are contiguous: b64 loads.
// ---------------------------------------------------------------------------
__global__ __launch_bounds__(128) void k_gproj(const float* __restrict__ qws,
                                               const float* __restrict__ Wk,
                                               float* __restrict__ gws) {
    const int c    = blockIdx.x;
    const int wave = threadIdx.x >> 5;
    const int lane = threadIdx.x & 31;
    const int n0   = (blockIdx.y * 4 + wave) * 16;
    const int klo  = (lane >> 4) << 1;
    const int mlo  = lane & 15;

    const float* q0  = qws + (size_t)c * QBLK;
    const float* a0p = q0 + (size_t)mlo * EMB + klo;
    const float* a1p = q0 + (size_t)(mlo + 16) * EMB + klo;
    const float* bp  = Wk + (size_t)c * WBLK + (size_t)(n0 + mlo) * EMB + klo;

    v8f acc0 = {}; v8f acc1 = {};
#pragma unroll 4
    for (int kk = 0; kk < EMB; kk += 4) {
        v2f a0 = *reinterpret_cast<const v2f*>(a0p);
        v2f a1 = *reinterpret_cast<const v2f*>(a1p);
        v2f b  = *reinterpret_cast<const v2f*>(bp);
        acc0 = wmma4(a0, b, acc0);
        acc1 = wmma4(a1, b, acc1);
        a0p += 4; a1p += 4; bp += 4;
    }

    float* g0 = gws + (size_t)c * QBLK;
    const int d   = n0 + mlo;
    const int bhi = (lane >> 4) * 8;
#pragma unroll
    for (int i = 0; i < 8; ++i) {
        g0[(size_t)(i + bhi) * EMB + d]      = acc0[i];
        g0[(size_t)(i + bhi + 16) * EMB + d] = acc1[i];
    }
}

// ---------------------------------------------------------------------------
// Kernel 3: per (c,b):  s[r] = (1/sqrt(512)) * sum_d ns[b][d][c+r]*g[c][b][d]
//           a = softmax(s);  zbar[c][b][d] = sum_r a[r]*ns[b][d][c+r]
// One wave per (c,b). grid (64, 32), 32 thr.
// ---------------------------------------------------------------------------
__global__ __launch_bounds__(32) void k_attn(const float* __restrict__ ns,
                                             const float* __restrict__ gws,
                                             float* __restrict__ zws) {
    const int c    = blockIdx.x;
    const int b    = blockIdx.y;
    const int lane = threadIdx.x;

    const float* g     = gws + ((size_t)c * BATCH + b) * EMB;
    const float* zbase = ns + (size_t)b * NSB + c;

    float s[RF];
#pragma unroll
    for (int r = 0; r < RF; ++r) s[r] = 0.0f;

    for (int d = lane; d < EMB; d += 32) {
        const float gv = g[d];
        const float* zp = zbase + (size_t)d * LCOLS;
#pragma unroll
        for (int r = 0; r < RF; ++r) s[r] += gv * zp[r];
    }

    const float inv_sqrt_d = 0.04419417382415922f; // 1/sqrt(512)
#pragma unroll
    for (int r = 0; r < RF; ++r) {
        float v = s[r];
        for (int off = 16; off > 0; off >>= 1) v += __shfl_xor(v, off, 32);
        s[r] = v * inv_sqrt_d;
    }

    float mx = s[0];
#pragma unroll
    for (int r = 1; r < RF; ++r) mx = fmaxf(mx, s[r]);
    float sum = 0.0f;
#pragma unroll
    for (int r = 0; r < RF; ++r) { s[r] = __expf(s[r] - mx); sum += s[r]; }
    const float inv = 1.0f / sum;
#pragma unroll
    for (int r = 0; r < RF; ++r) s[r] *= inv;

    float* z = zws + ((size_t)c * BATCH + b) * EMB;
    for (int d = lane; d < EMB; d += 32) {
        const float* zp = zbase + (size_t)d * LCOLS;
        float acc = 0.0f;
#pragma unroll
        for (int r = 0; r < RF; ++r) acc += s[r] * zp[r];
        z[d] = acc;
    }
}

// ---------------------------------------------------------------------------
// Kernel 4: y[c][b][e] = tanh( sum_d zbar[c][b][d] * Wv[c][d][e] )
//           out[b][e][c] = y[c][b][e]
// ---------------------------------------------------------------------------
__global__ __launch_bounds__(128) void k_out(const float* __restrict__ zws,
                                             const float* __restrict__ Wv,
                                             float* __restrict__ out) {
    const int c    = blockIdx.x;
    const int wave = threadIdx.x >> 5;
    const int lane = threadIdx.x & 31;
    const int n0   = (blockIdx.y * 4 + wave) * 16;
    const int klo  = (lane >> 4) << 1;
    const int mlo  = lane & 15;

    const float* z0  = zws + (size_t)c * QBLK;
    const float* a0p = z0 + (size_t)mlo * EMB + klo;
    const float* a1p = z0 + (size_t)(mlo + 16) * EMB + klo;
    const float* bp  = Wv + (size_t)c * WBLK + (size_t)klo * EMB + n0 + mlo;

    v8f acc0 = {}; v8f acc1 = {};
#pragma unroll 4
    for (int kk = 0; kk < EMB; kk += 4) {
        v2f a0 = *reinterpret_cast<const v2f*>(a0p);
        v2f a1 = *reinterpret_cast<const v2f*>(a1p);
        v2f b;  b.x = bp[0]; b.y = bp[EMB];
        acc0 = wmma4(a0, b, acc0);
        acc1 = wmma4(a1, b, acc1);
        a0p += 4; a1p += 4; bp += 4 * EMB;
    }

    const int e   = n0 + mlo;
    const int bhi = (lane >> 4) * 8;
#pragma unroll
    for (int i = 0; i < 8; ++i) {
        const int b0 = i + bhi;
        const int b1 = b0 + 16;
        out[(size_t)b0 * NSB + (size_t)e * LCOLS + c] = tanhf(acc0[i]);
        out[(size_t)b1 * NSB + (size_t)e * LCOLS + c] = tanhf(acc1[i]);
    }
}

// ---------------------------------------------------------------------------
extern "C" void kernel_launch(void* const* d_in, const int* in_sizes, int n_in,
                              void* d_out, int out_size, void* d_ws, size_t ws_size,
                              hipStream_t stream) {
    const int*   idx = (const int*)d_in[0];
    const float* ns  = (const float*)d_in[1];
    const float* emb = (const float*)d_in[2];
    const float* Wq  = (const float*)d_in[3];
    const float* Wk  = (const float*)d_in[4];
    const float* Wv  = (const float*)d_in[5];
    float* out = (float*)d_out;

    float* qws = (float*)d_ws;                    // 64*32*512 f32 = 4 MB
    float* gws = qws + (size_t)CONTEXT * QBLK;    // 4 MB
    float* zws = gws + (size_t)CONTEXT * QBLK;    // 4 MB

    const int ntot = BATCH * EMB * LCOLS;
    k_copy<<<(ntot + 255) / 256, 256, 0, stream>>>(ns, out, ntot);

    dim3 gg(CONTEXT, EMB / 64);  // (64, 8) blocks x 4 waves -> 16-wide N tiles
    k_qproj<<<gg, 128, 0, stream>>>(idx, ns, emb, Wq, qws);
    k_gproj<<<gg, 128, 0, stream>>>(qws, Wk, gws);
    k_attn <<<dim3(CONTEXT, BATCH), 32, 0, stream>>>(ns, gws, zws);
    k_out  <<<gg, 128, 0, stream>>>(zws, Wv, out);
}